// PositionalEncoding_78855599554887
// MI455X (gfx1250) — compile-verified
//
#include <hip/hip_runtime.h>

// out = (x + sinusoidal_pe(s, j)) * dropout_mask
// N=8, S=4096, D=1024, fp32. Memory-bound: ~402 MB traffic -> ~17us floor @ 23.3 TB/s.
// One block per s-row; 256 threads x float4 covers D=1024 exactly.
// PE computed once per thread (2x sincosf) and reused across all N batches.
// Iteration-0 loads issued BEFORE the transcendental block so HBM latency is
// hidden behind the PE math. b128 non-temporal loads/stores (single-touch
// stream > 192 MB L2).

#define S_DIM 4096
#define D_DIM 1024

typedef __attribute__((ext_vector_type(4))) float v4f;

__global__ __launch_bounds__(256) void
PositionalEncoding_78855599554887_kernel(const float* __restrict__ x,
                                         const float* __restrict__ mask,
                                         float* __restrict__ out,
                                         int N)
{
    const int s   = blockIdx.x;        // sequence position, 0..S-1
    const int tid = threadIdx.x;       // 0..255
    const int j0  = tid << 2;          // first of 4 contiguous D elements

    const size_t batchStride = (size_t)S_DIM * (size_t)D_DIM;
    size_t idx = (size_t)s * (size_t)D_DIM + (size_t)j0;

    // Kick off iteration 0's memory traffic before the transcendental block.
    v4f xv0 = __builtin_nontemporal_load((const v4f*)(x    + idx));
    v4f mv0 = __builtin_nontemporal_load((const v4f*)(mask + idx));

    // freq(j_even) = exp(j_even * (-ln(10000)/D)); pe even=sin, odd=cos
    const float c  = -9.210340371976184f / (float)D_DIM;   // -ln(10000)/D
    const float fs = (float)s;
    const float a0 = fs * expf((float)(j0)     * c);
    const float a2 = fs * expf((float)(j0 + 2) * c);

    float s0, c0, s2, c2;
    sincosf(a0, &s0, &c0);             // shared range reduction for the pair
    sincosf(a2, &s2, &c2);

    v4f pe;
    pe.x = s0;   // j0   even -> sin
    pe.y = c0;   // j0+1 odd  -> cos (same freq as j0)
    pe.z = s2;   // j0+2 even -> sin
    pe.w = c2;   // j0+3 odd  -> cos

    // Iteration 0 (loads already in flight).
    {
        v4f o;
        o.x = (xv0.x + pe.x) * mv0.x;
        o.y = (xv0.y + pe.y) * mv0.y;
        o.z = (xv0.z + pe.z) * mv0.z;
        o.w = (xv0.w + pe.w) * mv0.w;
        __builtin_nontemporal_store(o, (v4f*)(out + idx));
        idx += batchStride;
    }

#pragma unroll 7
    for (int n = 1; n < N; ++n, idx += batchStride) {
        v4f xv = __builtin_nontemporal_load((const v4f*)(x    + idx));
        v4f mv = __builtin_nontemporal_load((const v4f*)(mask + idx));
        v4f o;
        o.x = (xv.x + pe.x) * mv.x;
        o.y = (xv.y + pe.y) * mv.y;
        o.z = (xv.z + pe.z) * mv.z;
        o.w = (xv.w + pe.w) * mv.w;
        __builtin_nontemporal_store(o, (v4f*)(out + idx));
    }
}

extern "C" void kernel_launch(void* const* d_in, const int* in_sizes, int n_in,
                              void* d_out, int out_size, void* d_ws, size_t ws_size,
                              hipStream_t stream)
{
    const float* x    = (const float*)d_in[0];
    const float* mask = (const float*)d_in[1];
    float*       out  = (float*)d_out;

    const int N = in_sizes[0] / (S_DIM * D_DIM);   // = 8 for the reference shapes

    dim3 grid(S_DIM);       // one block per sequence position
    dim3 block(256);        // 256 threads * float4 = 1024 = D
    PositionalEncoding_78855599554887_kernel<<<grid, block, 0, stream>>>(x, mask, out, N);
}